// Covar_Attn_80066780332411
// MI455X (gfx1250) — compile-verified
//
#include <hip/hip_runtime.h>
#include <math.h>

// ---------------------------------------------------------------------------
// CDNA5 / gfx1250: covariance-pooling + Newton-Schulz sqrtm.
//  - All GEMMs through V_WMMA_F32_16X16X4_F32 (fp32 keeps the 14-matmul NS
//    chain numerically stable; ~133 GFLOP vs a 4.4us HBM floor -> fp32 free).
//  - NS GEMMs (square, no transpose) stage tiles with the Tensor Data Mover
//    (tensor_load_to_lds, TENSORcnt double buffering); the D# padding fields
//    reproduce the bank-conflict-free LDS strides (16->18, 128->132) in HW.
//  - cov GEMM needs a transpose during staging -> manual VALU staging path.
// ---------------------------------------------------------------------------

typedef float v2f __attribute__((ext_vector_type(2)));
typedef float v8f __attribute__((ext_vector_type(8)));
typedef unsigned int v4u __attribute__((ext_vector_type(4)));
typedef int v8i __attribute__((ext_vector_type(8)));
typedef int v4i __attribute__((ext_vector_type(4)));

#define TILE 128
#define KT   16

#define MODE_PLAIN 0
#define MODE_COV   1
#define MODE_NST   2   // d = 1.5*(i==j) - 0.5*acc

// LDS byte offset of a __shared__ object (generic ptr -> addrspace(3) -> int)
__device__ __forceinline__ unsigned lds_off(const void* p)
{
    return (unsigned)(unsigned long long)
        (__attribute__((address_space(3))) const void*)p;
}

// Issue one 2D TDM tile load: global (row-major, stride_el) -> LDS (+padding).
// D# layout per CDNA5 ISA ch.8: group0 = {flags, lds_addr, global_addr, type},
// group1 = {mask/data_size/pad, dims, tile dims, strides}. Tile == tensor so
// no OOB clamping is ever triggered.
__device__ __forceinline__ void tdm_load_2d(unsigned lds_addr,
                                            const float* gptr,
                                            unsigned tile_w,   // elements
                                            unsigned tile_h,   // rows
                                            unsigned stride_el,
                                            unsigned pad_interval, // code, 0=none
                                            unsigned pad_amount)   // code
{
    unsigned long long ga = (unsigned long long)(size_t)gptr;
    v4u g0;
    g0[0] = 1u;                                      // count=1, user descriptor
    g0[1] = lds_addr;                                // lds_addr [63:32]
    g0[2] = (unsigned)(ga & 0xFFFFFFFFull);          // global_addr[31:0]
    g0[3] = (unsigned)((ga >> 32) & 0x01FFFFFFull)   // global_addr[56:32]
          | (2u << 30);                              // type=2 ("image")
    unsigned pe = (pad_interval | pad_amount) ? 1u : 0u;
    v8i g1;
    g1[0] = (int)((2u << 16)                         // data_size = 4 bytes
          | (pe << 20)                               // pad_enable
          | (pad_interval << 22)                     // pad_interval[24:22]
          | (pad_amount << 25));                     // pad_amount[31:25]
    g1[1] = (int)((tile_w & 0xFFFFu) << 16);         // tensor_dim0[15:0]
    g1[2] = (int)((tile_w >> 16) | ((tile_h & 0xFFFFu) << 16)); // d0 hi | d1 lo
    g1[3] = (int)((tile_h >> 16) | (tile_w << 16));  // d1 hi | tile_dim0
    g1[4] = (int)tile_h;                             // tile_dim1 (tile_dim2=0)
    g1[5] = (int)stride_el;                          // tensor_dim0_stride lo
    g1[6] = 0;                                       // stride0 hi | stride1 lo
    g1[7] = 0;
    v4i z4 = {0, 0, 0, 0};
    v8i z8 = {0, 0, 0, 0, 0, 0, 0, 0};
    __builtin_amdgcn_tensor_load_to_lds(g0, g1, z4, z4, z8, 0);
}

// ---------------------------------------------------------------------------
// NS GEMM: D[b] = epilogue(A[b] * B[b]), all matrices 512x512 row-major.
// TDM-staged, double-buffered LDS.
// ---------------------------------------------------------------------------
template <int MODE>
__global__ __launch_bounds__(256)
void gemm512_wmma_tdm(const float* __restrict__ A,
                      const float* __restrict__ Bm,
                      float* __restrict__ D)
{
    __shared__ float As[2][TILE][18];       // TDM pad: 16 dw + 2 dw  -> 18
    __shared__ float Bs[2][KT][TILE + 4];   // TDM pad: 128 dw + 4 dw -> 132

    const int b = blockIdx.z;
    const int K = 512, ld = 512;
    const float* __restrict__ Ab = A  + (size_t)b * 262144;
    const float* __restrict__ Bb = Bm + (size_t)b * 262144;
    float* __restrict__ Db = D + (size_t)b * 262144;

    const int tid  = threadIdx.x;
    const int lane = tid & 31;
    const int wave = tid >> 5;
    const int wrow = wave >> 1;
    const int wcol = wave & 1;
    const int lh   = lane >> 4;
    const int ll   = lane & 15;
    const int m0 = blockIdx.y * TILE;
    const int n0 = blockIdx.x * TILE;

    const unsigned asA[2] = { lds_off(&As[0][0][0]), lds_off(&As[1][0][0]) };
    const unsigned asB[2] = { lds_off(&Bs[0][0][0]), lds_off(&Bs[1][0][0]) };

    v8f acc[2][4];
#pragma unroll
    for (int mi = 0; mi < 2; ++mi)
#pragma unroll
        for (int ni = 0; ni < 4; ++ni)
            acc[mi][ni] = (v8f)0.0f;

    // prologue: DMA tile 0 into buffer 0 (wave 0 only; TDM is per-wave)
    if (tid < 32) {
        tdm_load_2d(asA[0], Ab + (size_t)m0 * ld, KT, TILE, ld, 3u, 1u);
        tdm_load_2d(asB[0], Bb + n0,              TILE, KT, ld, 6u, 3u);
    }

    int buf = 0;
    for (int kt = 0; kt < K; kt += KT) {
        if (tid < 32) {
            if (kt + KT < K) {   // prefetch next k-tile into the other buffer
                const int kn = kt + KT;
                tdm_load_2d(asA[buf ^ 1], Ab + (size_t)m0 * ld + kn,
                            KT, TILE, ld, 3u, 1u);
                tdm_load_2d(asB[buf ^ 1], Bb + (size_t)kn * ld + n0,
                            TILE, KT, ld, 6u, 3u);
                __builtin_amdgcn_s_wait_tensorcnt(2);  // current tile landed
            } else {
                __builtin_amdgcn_s_wait_tensorcnt(0);
            }
        }
        __syncthreads();

#pragma unroll
        for (int kk = 0; kk < KT; kk += 4) {
            v2f afr[2];
#pragma unroll
            for (int mi = 0; mi < 2; ++mi) {
                int r = wrow * 32 + mi * 16 + ll;
                afr[mi] = *(const v2f*)&As[buf][r][kk + lh * 2];
            }
#pragma unroll
            for (int ni = 0; ni < 4; ++ni) {
                int c = wcol * 64 + ni * 16 + ll;
                v2f bfr;
                bfr.x = Bs[buf][kk + lh * 2 + 0][c];
                bfr.y = Bs[buf][kk + lh * 2 + 1][c];
#pragma unroll
                for (int mi = 0; mi < 2; ++mi) {
                    acc[mi][ni] = __builtin_amdgcn_wmma_f32_16x16x4_f32(
                        false, afr[mi], false, bfr,
                        (short)0, acc[mi][ni], false, false);
                }
            }
        }
        __syncthreads();   // everyone done with buf before it is re-DMA'd
        buf ^= 1;
    }

#pragma unroll
    for (int mi = 0; mi < 2; ++mi)
#pragma unroll
        for (int ni = 0; ni < 4; ++ni)
#pragma unroll
            for (int r = 0; r < 8; ++r) {
                int grow = m0 + wrow * 32 + mi * 16 + lh * 8 + r;
                int gcol = n0 + wcol * 64 + ni * 16 + ll;
                float v = acc[mi][ni][r];
                if (MODE == MODE_NST) v = (grow == gcol ? 1.5f : 0.0f) - 0.5f * v;
                Db[(size_t)grow * ld + gcol] = v;
            }
}

// ---------------------------------------------------------------------------
// cov GEMM: D[b] = (Xc[b] * Xc[b]^T) / M. B staging transposes -> manual path.
// ---------------------------------------------------------------------------
__global__ __launch_bounds__(256)
void gemm_cov_wmma(const float* __restrict__ X, float* __restrict__ D,
                   int K, int lda, float alpha)
{
    __shared__ float As[TILE][18];
    __shared__ float Bs[KT][TILE + 4];

    const int b = blockIdx.z;
    const float* __restrict__ Xb = X + (size_t)b * ((size_t)512 * lda);
    float* __restrict__ Db = D + (size_t)b * 262144;

    const int tid  = threadIdx.x;
    const int lane = tid & 31;
    const int wave = tid >> 5;
    const int wrow = wave >> 1;
    const int wcol = wave & 1;
    const int lh   = lane >> 4;
    const int ll   = lane & 15;
    const int m0 = blockIdx.y * TILE;
    const int n0 = blockIdx.x * TILE;

    v8f acc[2][4];
#pragma unroll
    for (int mi = 0; mi < 2; ++mi)
#pragma unroll
        for (int ni = 0; ni < 4; ++ni)
            acc[mi][ni] = (v8f)0.0f;

    for (int kt = 0; kt < K; kt += KT) {
#pragma unroll
        for (int p = 0; p < 2; ++p) {          // A tile 128x16
            int i  = tid + p * 256;
            int r  = i >> 2;
            int c4 = (i & 3) << 2;
            float4 v = *(const float4*)&Xb[(size_t)(m0 + r) * lda + kt + c4];
            As[r][c4 + 0] = v.x; As[r][c4 + 1] = v.y;
            As[r][c4 + 2] = v.z; As[r][c4 + 3] = v.w;
        }
#pragma unroll
        for (int p = 0; p < 2; ++p) {          // B tile = X^T, transpose in LDS
            int i  = tid + p * 256;
            int n  = i >> 2;
            int k4 = (i & 3) << 2;
            float4 v = *(const float4*)&Xb[(size_t)(n0 + n) * lda + kt + k4];
            Bs[k4 + 0][n] = v.x; Bs[k4 + 1][n] = v.y;
            Bs[k4 + 2][n] = v.z; Bs[k4 + 3][n] = v.w;
        }
        __syncthreads();

#pragma unroll
        for (int kk = 0; kk < KT; kk += 4) {
            v2f afr[2];
#pragma unroll
            for (int mi = 0; mi < 2; ++mi) {
                int r = wrow * 32 + mi * 16 + ll;
                afr[mi] = *(const v2f*)&As[r][kk + lh * 2];
            }
#pragma unroll
            for (int ni = 0; ni < 4; ++ni) {
                int c = wcol * 64 + ni * 16 + ll;
                v2f bfr;
                bfr.x = Bs[kk + lh * 2 + 0][c];
                bfr.y = Bs[kk + lh * 2 + 1][c];
#pragma unroll
                for (int mi = 0; mi < 2; ++mi) {
                    acc[mi][ni] = __builtin_amdgcn_wmma_f32_16x16x4_f32(
                        false, afr[mi], false, bfr,
                        (short)0, acc[mi][ni], false, false);
                }
            }
        }
        __syncthreads();
    }

#pragma unroll
    for (int mi = 0; mi < 2; ++mi)
#pragma unroll
        for (int ni = 0; ni < 4; ++ni)
#pragma unroll
            for (int r = 0; r < 8; ++r) {
                int grow = m0 + wrow * 32 + mi * 16 + lh * 8 + r;
                int gcol = n0 + wcol * 64 + ni * 16 + ll;
                Db[(size_t)grow * 512 + gcol] = acc[mi][ni][r] * alpha;
            }
}

// ---- x (B,C,M): subtract per-(b,c) mean over M ---------------------------
__global__ __launch_bounds__(256)
void center_kernel(const float* __restrict__ x, float* __restrict__ Xc)
{
    const int c = blockIdx.x, b = blockIdx.y;
    const float* __restrict__ row = x  + ((size_t)b * 512 + c) * 784;
    float* __restrict__ orow      = Xc + ((size_t)b * 512 + c) * 784;
    __shared__ float red[256];
    float s = 0.0f;
    for (int i = threadIdx.x; i < 784; i += 256) s += row[i];
    red[threadIdx.x] = s;
    __syncthreads();
    for (int o = 128; o > 0; o >>= 1) {
        if (threadIdx.x < o) red[threadIdx.x] += red[threadIdx.x + o];
        __syncthreads();
    }
    const float mean = red[0] * (1.0f / 784.0f);
    for (int i = threadIdx.x; i < 784; i += 256) orow[i] = row[i] - mean;
}

// ---- normA[b] = trace(cov[b]) --------------------------------------------
__global__ __launch_bounds__(256)
void trace_kernel(const float* __restrict__ cov, float* __restrict__ normA)
{
    const int b = blockIdx.x;
    __shared__ float red[256];
    float s = 0.0f;
    for (int i = threadIdx.x; i < 512; i += 256)
        s += cov[(size_t)b * 262144 + (size_t)i * 513];
    red[threadIdx.x] = s;
    __syncthreads();
    for (int o = 128; o > 0; o >>= 1) {
        if (threadIdx.x < o) red[threadIdx.x] += red[threadIdx.x + o];
        __syncthreads();
    }
    if (threadIdx.x == 0) normA[b] = red[0];
}

// ---- Y = cov/normA , Z = I -----------------------------------------------
__global__ __launch_bounds__(256)
void init_kernel(const float* __restrict__ cov, const float* __restrict__ normA,
                 float* __restrict__ Y, float* __restrict__ Z)
{
    const size_t idx = (size_t)blockIdx.x * 256 + threadIdx.x;
    const int b  = (int)(idx >> 18);
    const int rc = (int)(idx & 262143);
    const int i = rc >> 9, j = rc & 511;
    Y[idx] = cov[idx] / normA[b];
    Z[idx] = (i == j) ? 1.0f : 0.0f;
}

// ---- scale[b][c] = sqrt(normA[b])/512 * sum_i Ysqrt[b][i][c] -------------
__global__ __launch_bounds__(512)
void colmean_kernel(const float* __restrict__ Y, const float* __restrict__ normA,
                    float* __restrict__ scale)
{
    const int b = blockIdx.x, c = threadIdx.x;
    const float* __restrict__ Yb = Y + (size_t)b * 262144;
    float s = 0.0f;
    for (int i = 0; i < 512; ++i) s += Yb[(size_t)i * 512 + c];
    scale[b * 512 + c] = s * sqrtf(normA[b]) * (1.0f / 512.0f);
}

// ---- out = x * scale[b][c] -----------------------------------------------
__global__ __launch_bounds__(256)
void apply_kernel(const float* __restrict__ x, const float* __restrict__ scale,
                  float* __restrict__ out)
{
    const size_t idx = (size_t)blockIdx.x * 256 + threadIdx.x;
    const int ch = (int)(idx / 784);   // = b*512 + c
    out[idx] = x[idx] * scale[ch];
}

extern "C" void kernel_launch(void* const* d_in, const int* in_sizes, int n_in,
                              void* d_out, int out_size, void* d_ws, size_t ws_size,
                              hipStream_t stream)
{
    (void)in_sizes; (void)n_in; (void)out_size; (void)ws_size;
    const float* x = (const float*)d_in[0];
    float* out = (float*)d_out;

    const int B = 32, C = 512, M = 784, D = 512;
    const size_t DD = (size_t)D * D;          // 262144
    const size_t CM = (size_t)C * M;          // 401408

    float* Xc = (float*)d_ws;                 // B*C*M
    float* U  = Xc + (size_t)B * CM;          // B*D*D (cov, then gemm scratch)
    float* Y  = U  + (size_t)B * DD;          // B*D*D
    float* Z  = Y  + (size_t)B * DD;          // B*D*D
    float* T  = Z  + (size_t)B * DD;          // B*D*D
    float* normA = T + (size_t)B * DD;        // B
    float* scale = normA + B;                 // B*C

    const dim3 gblk(256);
    const dim3 ggrid(D / TILE, D / TILE, B);  // 4 x 4 x 32

    // 1) center
    center_kernel<<<dim3(C, B), 256, 0, stream>>>(x, Xc);

    // 2) cov = Xc * Xc^T / M   (K = 784, manual-transpose staging)
    gemm_cov_wmma<<<ggrid, gblk, 0, stream>>>(Xc, U, M, M, 1.0f / (float)M);

    // 3) trace + normalize, Z = I
    trace_kernel<<<B, 256, 0, stream>>>(U, normA);
    init_kernel<<<(unsigned)((size_t)B * DD / 256), 256, 0, stream>>>(U, normA, Y, Z);

    // 4) Newton-Schulz: 4 full iterations + final half step (14 TDM GEMMs)
    for (int it = 0; it < 5; ++it) {
        // T = 1.5I - 0.5 * Z*Y
        gemm512_wmma_tdm<MODE_NST><<<ggrid, gblk, 0, stream>>>(Z, Y, T);
        // Ynew = Y * T
        gemm512_wmma_tdm<MODE_PLAIN><<<ggrid, gblk, 0, stream>>>(Y, T, U);
        { float* t = Y; Y = U; U = t; }
        if (it < 4) {
            // Znew = T * Z
            gemm512_wmma_tdm<MODE_PLAIN><<<ggrid, gblk, 0, stream>>>(T, Z, U);
            { float* t = Z; Z = U; U = t; }
        }
    }

    // 5) per-channel scale = mean over rows * sqrt(trace)
    colmean_kernel<<<B, 512, 0, stream>>>(Y, normA, scale);

    // 6) out = scale * x
    apply_kernel<<<(unsigned)((size_t)B * CM / 256), 256, 0, stream>>>(x, scale, out);
}